// SpodNet_61692910239919
// MI455X (gfx1250) — compile-verified
//
#include <hip/hip_runtime.h>
#include <hip/hip_bf16.h>

// ---------------------------------------------------------------------------
// SpodNet on MI455X (gfx1250, wave32)
//   Phase 1: W = inv(Theta) via Newton-Schulz  X <- 2X - X*Theta*X
//            batched 256x256 FP32 GEMMs on V_WMMA_F32_16X16X4_F32,
//            K-chunks staged to LDS by the Tensor Data Mover (double-buffered)
//   Phase 2: sequential 256-column sweep with W held entirely in LDS,
//            TDM-loaded with pad_interval=256/pad_amount=1 => row stride 257
//            (bank-conflict-free), rank-2 updates, no inv_T11 materialized
// ---------------------------------------------------------------------------

#define PDIM   256
#define BATCH  64
#define NS_ITERS 16
#define GY_CONST 1.0f

typedef float v2f __attribute__((ext_vector_type(2)));
typedef float v8f __attribute__((ext_vector_type(8)));
typedef unsigned int v4u __attribute__((ext_vector_type(4)));
typedef int v8i __attribute__((ext_vector_type(8)));
typedef int v4i __attribute__((ext_vector_type(4)));

#if defined(__has_builtin)
#if __has_builtin(__builtin_amdgcn_tensor_load_to_lds)
#define SPOD_HAVE_TDM 1
#endif
#endif

#if defined(SPOD_HAVE_TDM)
// ---------------------------------------------------------------------------
// Issue one 2-D TDM tile load: global (row-major, row stride = stride elems)
// -> LDS at byte offset lds_off, optionally inserting 1 DWORD of padding
// every (2<<pad_interval_code) DWORDs (reproduces padded LDS tile strides).
// D# layout per CDNA5 ISA.
// This toolchain's builtin takes 6 args (extra int32x8 group before cpol).
// ---------------------------------------------------------------------------
__device__ inline void tdm_load_2d(unsigned int lds_off, const void* gsrc,
                                   unsigned int tile_w, unsigned int tile_h,
                                   unsigned int tensor_w, unsigned int tensor_h,
                                   unsigned int stride,
                                   unsigned int pad_interval_code, bool pad_en)
{
    const unsigned long long ga = (unsigned long long)(uintptr_t)gsrc;
    v4u g0;
    g0[0] = 1u;                                         // count=1, user D#
    g0[1] = lds_off;                                    // lds_addr (bytes)
    g0[2] = (unsigned int)(ga & 0xFFFFFFFFu);           // global_addr[31:0]
    g0[3] = (unsigned int)((ga >> 32) & 0x01FFFFFFu)    // global_addr[56:32]
          | (2u << 30);                                 // type=2 ("image")
    unsigned int w0 = (2u << 16);                       // data_size=2 -> 4B
    if (pad_en) w0 |= (1u << 20)                        // pad_enable
               | (pad_interval_code << 22)              // interval: 2<<code DW
               | (0u << 25);                            // pad_amount: 1 DWORD
    v8i g1;
    g1[0] = (int)w0;                                    // wg_mask=0 (no cluster)
    g1[1] = (int)((tensor_w & 0xFFFFu) << 16);          // tensor_dim0[15:0]
    g1[2] = (int)(((tensor_w >> 16) & 0xFFFFu)          // tensor_dim0[31:16]
          | ((tensor_h & 0xFFFFu) << 16));              // tensor_dim1[15:0]
    g1[3] = (int)(((tensor_h >> 16) & 0xFFFFu)          // tensor_dim1[31:16]
          | ((tile_w & 0xFFFFu) << 16));                // tile_dim0
    g1[4] = (int)(tile_h & 0xFFFFu);                    // tile_dim1 (tile_dim2=0)
    g1[5] = (int)stride;                                // tensor_dim0_stride[31:0]
    g1[6] = 0;                                          // stride hi, dim1_stride lo
    g1[7] = 0;
    const v4i z4 = {0, 0, 0, 0};                        // 2-D: groups 2/3 unused
    const v8i z8 = {0, 0, 0, 0, 0, 0, 0, 0};
    __builtin_amdgcn_tensor_load_to_lds(g0, g1, z4, z4, z8, 0);
}
__device__ inline unsigned int lds_offset_of(const void* p)
{
    // generic shared pointer = {aperture_hi, AS3 offset}; low 32 bits = offset
    return (unsigned int)(uintptr_t)p;
}
#endif

// ---------------------------------------------------------------------------
// alpha[b] = 1 / ||Theta_b||_inf   (row-sum norm; SPD => Newton-Schulz seed)
// ---------------------------------------------------------------------------
__global__ __launch_bounds__(256) void spod_alpha_kernel(
    const float* __restrict__ Theta, float* __restrict__ alpha)
{
    const int b = blockIdx.x;
    const int t = threadIdx.x;
    const float* Tb = Theta + (size_t)b * PDIM * PDIM;
    float s = 0.0f;
    #pragma unroll 8
    for (int j = 0; j < PDIM; ++j) s += fabsf(Tb[(size_t)t * PDIM + j]);
    __shared__ float red[PDIM];
    red[t] = s;
    __syncthreads();
    for (int off = PDIM / 2; off > 0; off >>= 1) {
        if (t < off) red[t] = fmaxf(red[t], red[t + off]);
        __syncthreads();
    }
    if (t == 0) alpha[b] = 1.0f / red[0];
}

// X0 = alpha[b] * I
__global__ __launch_bounds__(256) void spod_init_x(
    float* __restrict__ X, const float* __restrict__ alpha)
{
    size_t idx = (size_t)blockIdx.x * blockDim.x + threadIdx.x;
    int b  = (int)(idx >> 16);
    int ij = (int)(idx & 0xFFFF);
    int i = ij >> 8, j = ij & 255;
    X[idx] = (i == j) ? alpha[b] : 0.0f;
}

// ---------------------------------------------------------------------------
// Batched FP32 GEMM on WMMA 16x16x4, TDM-staged, double-buffered K chunks.
//   MODE 0:  C = A * B
//   MODE 1:  C = 2*Xold - A * B      (Newton-Schulz update, Xold == A)
// Block tile 128x64, 8 waves, wave w owns rows [16w,16w+16) x 64 cols.
// ---------------------------------------------------------------------------
template <int MODE>
__global__ __launch_bounds__(256) void spod_gemm(
    const float* __restrict__ Ab, const float* __restrict__ Bb,
    const float* __restrict__ Xb, float* __restrict__ Cb)
{
    __shared__ float As[2][128][33];  // stride 33: conflict-free fragment reads
    __shared__ float Bs[2][32][65];   // stride 65

    const int b = blockIdx.z;
    const float* A = Ab + (size_t)b * PDIM * PDIM;
    const float* B = Bb + (size_t)b * PDIM * PDIM;
    float*       C = Cb + (size_t)b * PDIM * PDIM;

    const int rowBase = blockIdx.y * 128;
    const int colBase = blockIdx.x * 64;
    const int tid  = threadIdx.x;
    const int wave = tid >> 5;        // 0..7
    const int lane = tid & 31;        // wave32
    const int mrow  = lane & 15;      // M (A/C) or N (B) within 16-tile
    const int khalf = (lane >> 4) << 1;  // K pair base: 0 (lanes 0-15), 2 (16-31)

    v8f acc[4];
    #pragma unroll
    for (int nt = 0; nt < 4; ++nt)
        #pragma unroll
        for (int e = 0; e < 8; ++e) acc[nt][e] = 0.0f;

#if defined(SPOD_HAVE_TDM)
    const bool issuer = (tid < 32);   // wave 0 drives the TDM
    const unsigned int asbase = lds_offset_of(&As[0][0][0]);
    const unsigned int bsbase = lds_offset_of(&Bs[0][0][0]);
    const unsigned int asbuf  = 128u * 33u * 4u;
    const unsigned int bsbuf  = 32u * 65u * 4u;
    if (issuer) {
        // chunk 0 -> buffer 0
        tdm_load_2d(asbase, A + (size_t)rowBase * PDIM,
                    32, 128, PDIM, PDIM, PDIM, /*interval 32DW*/4, true);
        tdm_load_2d(bsbase, B + colBase,
                    64, 32, PDIM, PDIM, PDIM, /*interval 64DW*/5, true);
        __builtin_amdgcn_s_wait_tensorcnt(0);
    }
    __syncthreads();
#endif

    for (int c = 0; c < PDIM / 32; ++c) {
        const int buf = c & 1;
#if defined(SPOD_HAVE_TDM)
        if (issuer && (c + 1) < PDIM / 32) {
            const int k0n = (c + 1) * 32;
            const unsigned int nbuf = (unsigned int)((c + 1) & 1);
            // async: DMA next chunk while this chunk is computed
            tdm_load_2d(asbase + nbuf * asbuf,
                        A + (size_t)rowBase * PDIM + k0n,
                        32, 128, PDIM, PDIM, PDIM, 4, true);
            tdm_load_2d(bsbase + nbuf * bsbuf,
                        B + (size_t)k0n * PDIM + colBase,
                        64, 32, PDIM, PDIM, PDIM, 5, true);
        }
#else
        {   // fallback: cooperative staged loads
            const int k0 = c * 32;
            const int r  = tid >> 1;
            const int kc = (tid & 1) * 16;
            const float* src = A + (size_t)(rowBase + r) * PDIM + k0 + kc;
            #pragma unroll
            for (int e = 0; e < 16; e += 4) {
                float4 v = *(const float4*)(src + e);
                As[buf][r][kc + e + 0] = v.x; As[buf][r][kc + e + 1] = v.y;
                As[buf][r][kc + e + 2] = v.z; As[buf][r][kc + e + 3] = v.w;
            }
            const int rb = tid >> 3;
            const int c0 = (tid & 7) * 8;
            const float* srcb = B + (size_t)(k0 + rb) * PDIM + colBase + c0;
            #pragma unroll
            for (int e = 0; e < 8; e += 4) {
                float4 v = *(const float4*)(srcb + e);
                Bs[buf][rb][c0 + e + 0] = v.x; Bs[buf][rb][c0 + e + 1] = v.y;
                Bs[buf][rb][c0 + e + 2] = v.z; Bs[buf][rb][c0 + e + 3] = v.w;
            }
            __syncthreads();
        }
#endif
        // --- 8 x (WMMA 16x16x4) steps over this K chunk ---
        #pragma unroll
        for (int kk = 0; kk < 32; kk += 4) {
            v2f afrag;  // A 16x4: lane -> row=lane%16, Ks {khalf, khalf+1}
            afrag.x = As[buf][wave * 16 + mrow][kk + khalf + 0];
            afrag.y = As[buf][wave * 16 + mrow][kk + khalf + 1];
            #pragma unroll
            for (int nt = 0; nt < 4; ++nt) {
                v2f bfrag;  // B 4x16: lane -> col=lane%16, Ks {khalf, khalf+1}
                bfrag.x = Bs[buf][kk + khalf + 0][nt * 16 + mrow];
                bfrag.y = Bs[buf][kk + khalf + 1][nt * 16 + mrow];
                acc[nt] = __builtin_amdgcn_wmma_f32_16x16x4_f32(
                    false, afrag, false, bfrag, (short)0, acc[nt],
                    false, false);
            }
        }
#if defined(SPOD_HAVE_TDM)
        if (issuer) __builtin_amdgcn_s_wait_tensorcnt(0);  // next chunk landed
#endif
        __syncthreads();
    }

    // --- store: C/D layout: VGPR e, lanes0-15 -> M=e, lanes16-31 -> M=8+e ---
    const float* Xold = (MODE == 1) ? (Xb + (size_t)b * PDIM * PDIM) : nullptr;
    #pragma unroll
    for (int nt = 0; nt < 4; ++nt) {
        #pragma unroll
        for (int e = 0; e < 8; ++e) {
            const int r = rowBase + wave * 16 + e + 8 * (lane >> 4);
            const int col = colBase + nt * 16 + (lane & 15);
            float val = acc[nt][e];
            if (MODE == 1) val = 2.0f * Xold[(size_t)r * PDIM + col] - val;
            C[(size_t)r * PDIM + col] = val;
        }
    }
}

// ---------------------------------------------------------------------------
// Column sweep. One workgroup per batch; W (256x256, row stride 257 for
// conflict-free banks) lives in LDS the whole time. Thread t owns row t.
// W is brought in by one TDM tile load with pad_interval=256DW/pad_amount=1DW.
// ---------------------------------------------------------------------------
__global__ __launch_bounds__(256) void spod_sweep(
    const float* __restrict__ Wg, const float* __restrict__ rv,
    float* __restrict__ ThetaOut)
{
    extern __shared__ float lds[];
    const int WST = PDIM + 1;             // 257: row stride (bank-conflict-free)
    float* W    = lds;                    // 256*257 floats
    float* tv   = lds + PDIM * WST;       // t_full vector (256)
    float* vv   = tv + PDIM;              // v = W*t, later w12 (256)
    float* red  = vv + PDIM;              // reduction buf, later w12n (256)
    float* scal = red + PDIM;             // broadcast scalars (8)

    const int b = blockIdx.x;
    const int t = threadIdx.x;
    const float* Wsrc = Wg + (size_t)b * PDIM * PDIM;
    float*       To   = ThetaOut + (size_t)b * PDIM * PDIM;

#if defined(SPOD_HAVE_TDM)
    if (t < 32) {
        // whole 256x256 W tile -> LDS with stride-257 padding, one DMA
        tdm_load_2d(lds_offset_of(W), Wsrc,
                    PDIM, PDIM, PDIM, PDIM, PDIM, /*interval 256DW*/7, true);
        __builtin_amdgcn_s_wait_tensorcnt(0);
    }
    __syncthreads();
#else
    #pragma unroll 4
    for (int j = 0; j < PDIM; j += 4) {
        float4 x = *(const float4*)(Wsrc + (size_t)t * PDIM + j);
        W[t * WST + j + 0] = x.x; W[t * WST + j + 1] = x.y;
        W[t * WST + j + 2] = x.z; W[t * WST + j + 3] = x.w;
    }
    __syncthreads();
#endif

    for (int col = 0; col < PDIM; ++col) {
        // t_full: rand_vecs row `col` scattered over indices != col
        const float tval = (t == col)
            ? 0.0f
            : rv[(size_t)col * (PDIM - 1) + t - (t > col ? 1 : 0)];
        tv[t] = tval;
        __syncthreads();

        // v = W * t_full   (thread t: row t, 256 MACs from LDS)
        float acc = 0.0f;
        const float* wr = &W[t * WST];
        #pragma unroll 8
        for (int j = 0; j < PDIM; ++j) acc = fmaf(wr[j], tv[j], acc);
        vv[t]  = acc;
        red[t] = acc * tval;
        __syncthreads();

        // s1 = t12' W11 t12 (tree reduce)
        for (int off = PDIM / 2; off > 0; off >>= 1) {
            if (t < off) red[t] += red[t + off];
            __syncthreads();
        }
        if (t == 0) {
            const float s1  = red[0];
            const float a   = vv[col];                  // w12 . t12
            const float w22 = W[(size_t)col * WST + col];
            scal[0] = a;
            scal[1] = w22;
            scal[2] = GY_CONST + s1;                    // t22_next
            scal[3] = GY_CONST + a * a / w22;           // 1/w22n
        }
        __syncthreads();
        const float a     = scal[0];
        const float w22   = scal[1];
        const float t22   = scal[2];
        const float rw22n = scal[3];                    // = 1/w22n
        const float w22n  = 1.0f / rw22n;

        // v2 = v - (a/w22) w12 ;  w12n = -w22n v2   (zero at col slot)
        const float w12_t  = W[t * WST + col];
        const float v2     = vv[t] - (a / w22) * w12_t;
        const float w12n_t = (t == col) ? 0.0f : (-w22n * v2);
        red[t] = w12n_t;                                // w12n vector
        vv[t]  = (t == col) ? 0.0f : w12_t;             // old w12 vector
        __syncthreads();

        // rank-2 update of row t:  W += -(1/w22) w12 w12' + (1/w22n) w12n w12n'
        if (t != col) {
            const float cj1 = -(1.0f / w22) * w12_t;
            const float cj2 = rw22n * w12n_t;
            float* wrw = &W[t * WST];
            #pragma unroll 8
            for (int k = 0; k < PDIM; ++k)
                wrw[k] = fmaf(cj1, vv[k], fmaf(cj2, red[k], wrw[k]));
            wrw[col] = w12n_t;                          // new W[t,col]
        } else {
            float* wrw = &W[(size_t)col * WST];
            for (int k = 0; k < PDIM; ++k) wrw[k] = red[k];  // new row col
            wrw[col] = w22n;
        }

        // Theta column/row writes (global; L2-resident)
        To[(size_t)t * PDIM + col] = (t == col) ? t22 : tval;
        To[(size_t)col * PDIM + t] = (t == col) ? t22 : tval;
        __syncthreads();
    }
}

// ---------------------------------------------------------------------------
extern "C" void kernel_launch(void* const* d_in, const int* in_sizes, int n_in,
                              void* d_out, int out_size, void* d_ws, size_t ws_size,
                              hipStream_t stream)
{
    (void)in_sizes; (void)n_in; (void)out_size; (void)ws_size;
    const float* Theta = (const float*)d_in[0];
    const float* rv    = (const float*)d_in[1];
    float*       Tout  = (float*)d_out;

    const size_t MAT = (size_t)BATCH * PDIM * PDIM;       // floats per buffer
    char* ws = (char*)d_ws;
    float* X0    = (float*)(ws + 0 * MAT * sizeof(float));
    float* X1    = (float*)(ws + 1 * MAT * sizeof(float));
    float* Y     = (float*)(ws + 2 * MAT * sizeof(float));
    float* alpha = (float*)(ws + 3 * MAT * sizeof(float));

    // Phase 1: W = inv(Theta) by Newton-Schulz on WMMA GEMMs
    spod_alpha_kernel<<<BATCH, 256, 0, stream>>>(Theta, alpha);
    spod_init_x<<<(unsigned)(MAT / 256), 256, 0, stream>>>(X0, alpha);

    dim3 ggrid(PDIM / 64, PDIM / 128, BATCH);             // (4, 2, 64)
    float* Xc = X0;
    float* Xn = X1;
    for (int it = 0; it < NS_ITERS; ++it) {
        spod_gemm<0><<<ggrid, 256, 0, stream>>>(Theta, Xc, nullptr, Y); // Y = T*X
        spod_gemm<1><<<ggrid, 256, 0, stream>>>(Xc, Y, Xc, Xn);         // Xn = 2X - X*Y
        float* tmp = Xc; Xc = Xn; Xn = tmp;
    }

    // Phase 2: LDS-resident column sweep (257-stride W + buffers per workgroup)
    const size_t sweep_lds = ((size_t)PDIM * (PDIM + 1) + 3 * PDIM + 8) * sizeof(float);
    spod_sweep<<<BATCH, 256, sweep_lds, stream>>>(Xc, rv, Tout);
}